// CombineGraph_56238301774297
// MI455X (gfx1250) — compile-verified
//
#include <hip/hip_runtime.h>
#include <math.h>

// ---------------------------------------------------------------------------
// CombineGraph on MI455X (gfx1250): fp32 WMMA (V_WMMA_F32_16X16X4_F32) for all
// GEMM-shaped work; wave32, 8 waves / 256-thread block, 64x64 output tiles.
// ---------------------------------------------------------------------------

typedef __attribute__((ext_vector_type(2))) float v2f;
typedef __attribute__((ext_vector_type(8))) float v8f;

#define BATCH 128
#define NNODE 64
#define DDIM  256
#define BN    (BATCH * NNODE)        // 8192
#define SZ    ((long)BN * DDIM)      // 2,097,152 floats per [B,N,D] tensor

__device__ __forceinline__ v8f wmma4(v2f a, v2f b, v8f c) {
  // D = A(16x4,f32) * B(4x16,f32) + C(16x16,f32); exact fp32 matrix op
  return __builtin_amdgcn_wmma_f32_16x16x4_f32(
      false, a, false, b, (short)0, c, false, false);
}

__device__ __forceinline__ float lrelu(float x) {
  return x > 0.0f ? x : 0.2f * x;
}

// ---------------------------------------------------------------------------
// h0[bn,:] = embedding[inputs[bn],:]
// ---------------------------------------------------------------------------
__global__ __launch_bounds__(64) void gather_kernel(
    const float* __restrict__ emb, const int* __restrict__ idx,
    float* __restrict__ out) {
  int bn = blockIdx.x;
  int t = threadIdx.x;                       // 64 threads * float4 = 256 floats
  long src = (long)idx[bn] * DDIM + t * 4;
  float4 v = *(const float4*)(emb + src);
  *(float4*)(out + (long)bn * DDIM + t * 4) = v;
}

// ---------------------------------------------------------------------------
// sess[b,d] = sum_n emb[item[b,n],d]*mask[b,n] / sum_n mask[b,n]
// ---------------------------------------------------------------------------
__global__ __launch_bounds__(256) void sess_mean_kernel(
    const float* __restrict__ emb, const int* __restrict__ item,
    const int* __restrict__ mask, float* __restrict__ sess) {
  int b = blockIdx.x;
  int d = threadIdx.x;
  float acc = 0.0f, cnt = 0.0f;
  for (int n = 0; n < NNODE; ++n) {
    float m = (float)mask[b * NNODE + n];
    cnt += m;
    acc += m * emb[(long)item[b * NNODE + n] * DDIM + d];
  }
  sess[b * DDIM + d] = acc / cnt;
}

// ---------------------------------------------------------------------------
// sbias[j,b,d] = b_g[j,d] + sum_k sess[b,k] * w_g[j, D+k, d]
// ---------------------------------------------------------------------------
__global__ __launch_bounds__(256) void sess_bias_kernel(
    const float* __restrict__ sess, const float* __restrict__ w_g,
    const float* __restrict__ b_g, float* __restrict__ sbias) {
  int b = blockIdx.x, j = blockIdx.y, d = threadIdx.x;
  __shared__ float ss[DDIM];
  ss[d] = sess[b * DDIM + d];
  __syncthreads();
  const float* W2 = w_g + (long)j * 2 * DDIM * DDIM + (long)DDIM * DDIM;
  float acc = b_g[j * DDIM + d];
  for (int k = 0; k < DDIM; ++k) acc += ss[k] * W2[(long)k * DDIM + d];
  sbias[((long)j * BATCH + b) * DDIM + d] = acc;
}

// ---------------------------------------------------------------------------
// Local aggregator, one batch per block.
//  e_k[i,j] = sum_d h[i,d]*a_k[d]*h[j,d]  (k=0..3)  -> WMMA, 4 accumulators
//  att = select-by-adj(leakyrelu(e_k), -9e15); softmax rows; out = att @ h
// ---------------------------------------------------------------------------
__global__ __launch_bounds__(256) void local_agg_kernel(
    const float* __restrict__ h_in, const int* __restrict__ adj,
    const float* __restrict__ a_loc, float* __restrict__ h_out) {
  __shared__ float hs[64][68];
  __shared__ float atts[64][68];
  __shared__ float aw[4][DDIM];

  int b = blockIdx.x;
  int tid = threadIdx.x;
  int lane = tid & 31, wv = tid >> 5;
  int lm = lane & 15, half = lane >> 4, kb = half * 2;
  const float* hb = h_in + (long)b * NNODE * DDIM;
  const int* adjb = adj + (long)b * NNODE * NNODE;

  for (int i = tid; i < 4 * DDIM; i += 256) aw[i >> 8][i & 255] = a_loc[i];

  v8f acc[2][4] = {};

  // ---- phase 1: e_k over D in 4 chunks of 64 ----
  for (int c = 0; c < 4; ++c) {
    int r = tid >> 2, s = (tid & 3) * 16;
    const float* src = hb + (long)r * DDIM + c * 64 + s;
#pragma unroll
    for (int i = 0; i < 16; i += 4) {
      float4 f = *(const float4*)(src + i);
      hs[r][s + i] = f.x; hs[r][s + i + 1] = f.y;
      hs[r][s + i + 2] = f.z; hs[r][s + i + 3] = f.w;
    }
    __syncthreads();
#pragma unroll
    for (int t = 0; t < 2; ++t) {
      int tile = wv * 2 + t, mt = tile >> 2, nt = tile & 3;
#pragma unroll
      for (int kk = 0; kk < 64; kk += 4) {
        float ra0 = hs[mt * 16 + lm][kk + kb];
        float ra1 = hs[mt * 16 + lm][kk + kb + 1];
        v2f bv; bv.x = hs[nt * 16 + lm][kk + kb];
                bv.y = hs[nt * 16 + lm][kk + kb + 1];
        int d0 = c * 64 + kk + kb;
#pragma unroll
        for (int k = 0; k < 4; ++k) {
          v2f av; av.x = ra0 * aw[k][d0]; av.y = ra1 * aw[k][d0 + 1];
          acc[t][k] = wmma4(av, bv, acc[t][k]);
        }
      }
    }
    __syncthreads();
  }

  // ---- att assembly: leaky-relu + adj select ----
#pragma unroll
  for (int t = 0; t < 2; ++t) {
    int tile = wv * 2 + t, mt = tile >> 2, nt = tile & 3;
#pragma unroll
    for (int v = 0; v < 8; ++v) {
      int m = mt * 16 + v + 8 * half;
      int n = nt * 16 + lm;
      int a4 = adjb[m * NNODE + n];
      float e0 = lrelu(acc[t][0][v]);
      float e1 = lrelu(acc[t][1][v]);
      float e2 = lrelu(acc[t][2][v]);
      float e3 = lrelu(acc[t][3][v]);
      float rr = -9e15f;
      rr = (a4 == 1) ? e0 : rr;
      rr = (a4 == 2) ? e1 : rr;
      rr = (a4 == 3) ? e2 : rr;
      rr = (a4 == 4) ? e3 : rr;
      atts[m][n] = rr;
    }
  }
  __syncthreads();

  // ---- row softmax ----
  if (tid < 64) {
    float mx = -3.402823466e38f;
    for (int j = 0; j < 64; ++j) mx = fmaxf(mx, atts[tid][j]);
    float sum = 0.0f;
    for (int j = 0; j < 64; ++j) {
      float e = __expf(atts[tid][j] - mx);
      atts[tid][j] = e;
      sum += e;
    }
    float inv = 1.0f / sum;
    for (int j = 0; j < 64; ++j) atts[tid][j] *= inv;
  }
  __syncthreads();

  // ---- phase 2: out = att @ h, D in 4 chunks ----
  float* ob = h_out + (long)b * NNODE * DDIM;
  for (int c = 0; c < 4; ++c) {
    int r = tid >> 2, s = (tid & 3) * 16;
    const float* src = hb + (long)r * DDIM + c * 64 + s;
#pragma unroll
    for (int i = 0; i < 16; i += 4) {
      float4 f = *(const float4*)(src + i);
      hs[r][s + i] = f.x; hs[r][s + i + 1] = f.y;
      hs[r][s + i + 2] = f.z; hs[r][s + i + 3] = f.w;
    }
    __syncthreads();
#pragma unroll
    for (int t = 0; t < 2; ++t) {
      int tile = wv * 2 + t, mt = tile >> 2, nt = tile & 3;
      v8f o = {};
#pragma unroll
      for (int kk = 0; kk < 64; kk += 4) {
        v2f av; av.x = atts[mt * 16 + lm][kk + kb];
                av.y = atts[mt * 16 + lm][kk + kb + 1];
        v2f bv; bv.x = hs[kk + kb][nt * 16 + lm];
                bv.y = hs[kk + kb + 1][nt * 16 + lm];
        o = wmma4(av, bv, o);
      }
#pragma unroll
      for (int v = 0; v < 8; ++v) {
        int m = mt * 16 + v + 8 * half;
        ob[(long)m * DDIM + c * 64 + nt * 16 + lm] = o[v];
      }
    }
    __syncthreads();
  }
}

// ---------------------------------------------------------------------------
// neigh[b] = hg_adj[b] @ h[b]   ([64,64] x [64,256]) per block
// ---------------------------------------------------------------------------
__global__ __launch_bounds__(256) void neigh_mm_kernel(
    const float* __restrict__ hg_adj, const float* __restrict__ h_in,
    float* __restrict__ out) {
  __shared__ float adjs[64][68];
  __shared__ float hs[64][68];
  int b = blockIdx.x;
  int tid = threadIdx.x;
  int lane = tid & 31, wv = tid >> 5;
  int lm = lane & 15, half = lane >> 4, kb = half * 2;
  const float* ab = hg_adj + (long)b * NNODE * NNODE;
  const float* hb = h_in + (long)b * NNODE * DDIM;
  float* ob = out + (long)b * NNODE * DDIM;

  {
    int r = tid >> 2, s = (tid & 3) * 16;
    const float* src = ab + (long)r * NNODE + s;
#pragma unroll
    for (int i = 0; i < 16; i += 4) {
      float4 f = *(const float4*)(src + i);
      adjs[r][s + i] = f.x; adjs[r][s + i + 1] = f.y;
      adjs[r][s + i + 2] = f.z; adjs[r][s + i + 3] = f.w;
    }
  }
  for (int c = 0; c < 4; ++c) {
    int r = tid >> 2, s = (tid & 3) * 16;
    const float* src = hb + (long)r * DDIM + c * 64 + s;
#pragma unroll
    for (int i = 0; i < 16; i += 4) {
      float4 f = *(const float4*)(src + i);
      hs[r][s + i] = f.x; hs[r][s + i + 1] = f.y;
      hs[r][s + i + 2] = f.z; hs[r][s + i + 3] = f.w;
    }
    __syncthreads();
#pragma unroll
    for (int t = 0; t < 2; ++t) {
      int tile = wv * 2 + t, mt = tile >> 2, nt = tile & 3;
      v8f o = {};
#pragma unroll
      for (int kk = 0; kk < 64; kk += 4) {
        v2f av; av.x = adjs[mt * 16 + lm][kk + kb];
                av.y = adjs[mt * 16 + lm][kk + kb + 1];
        v2f bv; bv.x = hs[kk + kb][nt * 16 + lm];
                bv.y = hs[kk + kb + 1][nt * 16 + lm];
        o = wmma4(av, bv, o);
      }
#pragma unroll
      for (int v = 0; v < 8; ++v) {
        int m = mt * 16 + v + 8 * half;
        ob[(long)m * DDIM + c * 64 + nt * 16 + lm] = o[v];
      }
    }
    __syncthreads();
  }
}

// ---------------------------------------------------------------------------
// General fp32 WMMA GEMM: C[M,N] = A[M,K] @ B + epilogue
//   B element (k,n) at Bp[k*bSK + n*bSN]  (handles W and W^T layouts)
//   biasMode: 0 none, 1 bias[n], 2 bias[(m>>6)*N + n] (per-batch sess bias)
//   act: 0 none, 1 relu
// ---------------------------------------------------------------------------
__global__ __launch_bounds__(256) void gemm_wmma_kernel(
    const float* __restrict__ A, const float* __restrict__ Bp,
    float* __restrict__ C, int M, int N, int K, long bSK, long bSN,
    const float* __restrict__ bias, int biasMode, int act) {
  __shared__ float As[64][68];
  __shared__ float Bs[64][68];
  int m0 = blockIdx.x * 64, n0 = blockIdx.y * 64;
  int tid = threadIdx.x;
  int lane = tid & 31, wv = tid >> 5;
  int lm = lane & 15, half = lane >> 4, kb = half * 2;

  v8f acc[2] = {};

  for (int kc = 0; kc < K; kc += 64) {
    int r = tid >> 2, s = (tid & 3) * 16;
    {
      const float* src = A + (long)(m0 + r) * K + kc + s;
      if (kc + 64 < K) __builtin_prefetch(src + 64, 0, 1);
#pragma unroll
      for (int i = 0; i < 16; i += 4) {
        float4 f = *(const float4*)(src + i);
        As[r][s + i] = f.x; As[r][s + i + 1] = f.y;
        As[r][s + i + 2] = f.z; As[r][s + i + 3] = f.w;
      }
    }
    {
      const float* src = Bp + (long)(kc + r) * bSK + (long)(n0 + s) * bSN;
#pragma unroll
      for (int i = 0; i < 16; ++i) Bs[r][s + i] = src[(long)i * bSN];
    }
    __syncthreads();
#pragma unroll
    for (int t = 0; t < 2; ++t) {
      int tile = wv * 2 + t, mt = tile >> 2, nt = tile & 3;
#pragma unroll
      for (int kk = 0; kk < 64; kk += 4) {
        v2f av; av.x = As[mt * 16 + lm][kk + kb];
                av.y = As[mt * 16 + lm][kk + kb + 1];
        v2f bv; bv.x = Bs[kk + kb][nt * 16 + lm];
                bv.y = Bs[kk + kb + 1][nt * 16 + lm];
        acc[t] = wmma4(av, bv, acc[t]);
      }
    }
    __syncthreads();
  }

#pragma unroll
  for (int t = 0; t < 2; ++t) {
    int tile = wv * 2 + t, mt = tile >> 2, nt = tile & 3;
    int n = n0 + nt * 16 + lm;
#pragma unroll
    for (int v = 0; v < 8; ++v) {
      int m = m0 + mt * 16 + v + 8 * half;
      float rr = acc[t][v];
      if (biasMode == 1) rr += bias[n];
      else if (biasMode == 2) rr += bias[(long)(m >> 6) * N + n];
      if (act == 1) rr = fmaxf(rr, 0.0f);
      C[(long)m * N + n] = rr;
    }
  }
}

// ---------------------------------------------------------------------------
// GRU gate fusion: out = (1-z)*n + z*h
// ---------------------------------------------------------------------------
__global__ __launch_bounds__(256) void gru_fuse_kernel(
    const float* __restrict__ gi, const float* __restrict__ gh,
    const float* __restrict__ hp, float* __restrict__ out) {
  long idx = (long)blockIdx.x * blockDim.x + threadIdx.x;
  if (idx >= SZ) return;
  long row = idx >> 8;
  int d = (int)(idx & 255);
  const float* gir = gi + row * 768;
  const float* ghr = gh + row * 768;
  float ir = gir[d], iz = gir[256 + d], in_ = gir[512 + d];
  float hr = ghr[d], hz = ghr[256 + d], hn = ghr[512 + d];
  float r = 1.0f / (1.0f + __expf(-(ir + hr)));
  float z = 1.0f / (1.0f + __expf(-(iz + hz)));
  float n = tanhf(in_ + r * hn);
  out[idx] = (1.0f - z) * n + z * hp[idx];
}

// ---------------------------------------------------------------------------
// final: h_local = 0.5*l_res + hl0 ; h_global = 0.5*g_res + hg1
//        d_out = [output | h_local | h_global]
// ---------------------------------------------------------------------------
__global__ __launch_bounds__(256) void combine_kernel(
    const float* __restrict__ hl0, const float* __restrict__ hg1,
    const float* __restrict__ l_res, const float* __restrict__ g_res,
    float* __restrict__ out) {
  long idx = (long)blockIdx.x * blockDim.x + threadIdx.x;
  if (idx >= SZ) return;
  float hloc = 0.5f * l_res[idx] + hl0[idx];
  float hglob = 0.5f * g_res[idx] + hg1[idx];
  out[idx] = hloc + hglob;
  out[SZ + idx] = hloc;
  out[2 * SZ + idx] = hglob;
}

// ---------------------------------------------------------------------------
extern "C" void kernel_launch(void* const* d_in, const int* in_sizes, int n_in,
                              void* d_out, int out_size, void* d_ws,
                              size_t ws_size, hipStream_t stream) {
  const int*   inputs = (const int*)d_in[0];
  const int*   adj    = (const int*)d_in[1];
  const int*   mask   = (const int*)d_in[2];
  const int*   item   = (const int*)d_in[3];
  const float* hg_adj = (const float*)d_in[4];
  const float* emb    = (const float*)d_in[5];
  const float* a_loc  = (const float*)d_in[6];
  const float* w_g    = (const float*)d_in[7];
  const float* b_g    = (const float*)d_in[8];
  const float* Wih    = (const float*)d_in[9];
  const float* Whh    = (const float*)d_in[10];
  const float* bih    = (const float*)d_in[11];
  const float* bhh    = (const float*)d_in[12];
  float* out = (float*)d_out;

  // workspace layout (floats), ~105 MiB total
  float* w    = (float*)d_ws;
  float* h0   = w;
  float* hl0  = h0 + SZ;
  float* hl1  = hl0 + SZ;
  float* hg0  = hl1 + SZ;
  float* hg1  = hg0 + SZ;
  float* gres = hg1 + SZ;
  float* lres = gres + SZ;
  float* gi   = lres + SZ;             // 8192*768
  float* gh   = gi + (long)BN * 768;
  float* neigh = gi;                   // alias: neigh dead before GRU phase
  float* sess = gh + (long)BN * 768;
  float* sb   = sess + BATCH * DDIM;   // sbias[2][B][D]

  dim3 blk(256);
  dim3 gElem((unsigned)((SZ + 255) / 256));
  dim3 gGemm256(BN / 64, DDIM / 64);   // (128, 4)
  dim3 gGemm768(BN / 64, 768 / 64);    // (128, 12)
  long wgStride = (long)2 * DDIM * DDIM;

  // embedding gathers + session mean
  gather_kernel<<<BN, 64, 0, stream>>>(emb, inputs, h0);
  sess_mean_kernel<<<BATCH, blk, 0, stream>>>(emb, item, mask, sess);
  sess_bias_kernel<<<dim3(BATCH, 2), blk, 0, stream>>>(sess, w_g, b_g, sb);

  // local hops
  local_agg_kernel<<<BATCH, blk, 0, stream>>>(h0, adj, a_loc, hl0);
  local_agg_kernel<<<BATCH, blk, 0, stream>>>(hl0, adj, a_loc + 4 * DDIM, hl1);

  // global hop 0: neigh = hg_adj@h0 ; hg0 = relu(neigh @ W0[:D] + sbias0)
  neigh_mm_kernel<<<BATCH, blk, 0, stream>>>(hg_adj, h0, neigh);
  gemm_wmma_kernel<<<gGemm256, blk, 0, stream>>>(
      neigh, w_g, hg0, BN, DDIM, DDIM, (long)DDIM, 1L, sb, 2, 1);
  // global hop 1
  neigh_mm_kernel<<<BATCH, blk, 0, stream>>>(hg_adj, hg0, neigh);
  gemm_wmma_kernel<<<gGemm256, blk, 0, stream>>>(
      neigh, w_g + wgStride, hg1, BN, DDIM, DDIM, (long)DDIM, 1L,
      sb + (long)BATCH * DDIM, 2, 1);

  // GRU 1: g_res = gru(x=hg0, h=hl0)
  gemm_wmma_kernel<<<gGemm768, blk, 0, stream>>>(
      hg0, Wih, gi, BN, 768, DDIM, 1L, (long)DDIM, bih, 1, 0);
  gemm_wmma_kernel<<<gGemm768, blk, 0, stream>>>(
      hl0, Whh, gh, BN, 768, DDIM, 1L, (long)DDIM, bhh, 1, 0);
  gru_fuse_kernel<<<gElem, blk, 0, stream>>>(gi, gh, hl0, gres);

  // GRU 2: l_res = gru(x=hl0, h=hg0)
  gemm_wmma_kernel<<<gGemm768, blk, 0, stream>>>(
      hl0, Wih, gi, BN, 768, DDIM, 1L, (long)DDIM, bih, 1, 0);
  gemm_wmma_kernel<<<gGemm768, blk, 0, stream>>>(
      hg0, Whh, gh, BN, 768, DDIM, 1L, (long)DDIM, bhh, 1, 0);
  gru_fuse_kernel<<<gElem, blk, 0, stream>>>(gi, gh, hg0, lres);

  // GRU 3: g_res = gru(x=g_res, h=hl1)
  gemm_wmma_kernel<<<gGemm768, blk, 0, stream>>>(
      gres, Wih, gi, BN, 768, DDIM, 1L, (long)DDIM, bih, 1, 0);
  gemm_wmma_kernel<<<gGemm768, blk, 0, stream>>>(
      hl1, Whh, gh, BN, 768, DDIM, 1L, (long)DDIM, bhh, 1, 0);
  gru_fuse_kernel<<<gElem, blk, 0, stream>>>(gi, gh, hl1, gres);

  // GRU 4: l_res = gru(x=l_res, h=hg1)
  gemm_wmma_kernel<<<gGemm768, blk, 0, stream>>>(
      lres, Wih, gi, BN, 768, DDIM, 1L, (long)DDIM, bih, 1, 0);
  gemm_wmma_kernel<<<gGemm768, blk, 0, stream>>>(
      hg1, Whh, gh, BN, 768, DDIM, 1L, (long)DDIM, bhh, 1, 0);
  gru_fuse_kernel<<<gElem, blk, 0, stream>>>(gi, gh, hg1, lres);

  // outputs
  combine_kernel<<<gElem, blk, 0, stream>>>(hl0, hg1, lres, gres, out);

  (void)in_sizes; (void)n_in; (void)out_size; (void)ws_size;
}